// BandSplit_27376121545519
// MI455X (gfx1250) — compile-verified
//
#include <hip/hip_runtime.h>

// BandSplit (BSRNN) for gfx1250: per-band LayerNorm + Linear(d->128) via
// V_WMMA_F32_16X16X4_F32. Store-bandwidth-bound (381 MB out @ 23.3 TB/s
// ~18 us floor); fp32 WMMA keeps exact precision at negligible compute cost.
// LDS layouts chosen so every WMMA fragment is a single aligned ds_load_b64:
//  - A: row stride 58 (even -> 8B aligned; 58 = -6 mod 64 -> conflict-free)
//  - W: K-pair interleaved, pair stride 288 (mod 64 = 32 -> hi/lo lane halves
//    hit opposite bank halves -> conflict-free)

typedef __attribute__((ext_vector_type(2))) float v2f;
typedef __attribute__((ext_vector_type(8))) float v8f;

#define TBLK        128    // time rows per workgroup
#define C_OUT       128
#define A_STRIDE    58     // even -> b64-aligned A fragments, conflict-free
#define WPAIR_STRIDE 288   // floats per K-pair row: 2*128 + 32
#define DPAD_MAX    56
#define T_LEN       3000
#define NBANDS      31
#define F_BINS      257
#define LN_EPS      1e-5f

__global__ __launch_bounds__(256) void bandsplit_wmma_kernel(
    const float* __restrict__ x,
    const float* __restrict__ g0_gamma, const float* __restrict__ g0_beta,
    const float* __restrict__ g0_W,     const float* __restrict__ g0_b,
    const float* __restrict__ g1_gamma, const float* __restrict__ g1_beta,
    const float* __restrict__ g1_W,     const float* __restrict__ g1_b,
    const float* __restrict__ g2_gamma, const float* __restrict__ g2_beta,
    const float* __restrict__ g2_W,     const float* __restrict__ g2_b,
    const float* __restrict__ g3_gamma, const float* __restrict__ g3_beta,
    const float* __restrict__ g3_W,     const float* __restrict__ g3_b,
    float* __restrict__ out)
{
    __shared__ float A_lds[TBLK * A_STRIDE];                  // normalized activations
    __shared__ float W_lds[(DPAD_MAX / 2) * WPAIR_STRIDE];    // weights, K-pair interleaved
    __shared__ float gamma_lds[DPAD_MAX];
    __shared__ float beta_lds[DPAD_MAX];
    __shared__ float mu_lds[TBLK];
    __shared__ float rstd_lds[TBLK];
    __shared__ float bias_lds[C_OUT];

    const int k      = blockIdx.x;          // band 0..30
    const int t_base = blockIdx.y * TBLK;   // time tile
    const int bb     = blockIdx.z;          // batch
    const int tid    = threadIdx.x;

    // ---- band metadata (GROUPS = [(10,3),(12,6),(8,16),(1,27)]) ----
    int bw, f0, idx;
    const float *gamma, *beta, *W, *bvec;
    if (k < 10)      { bw = 3;  idx = k;      f0 = idx * 3;        gamma = g0_gamma; beta = g0_beta; W = g0_W; bvec = g0_b; }
    else if (k < 22) { bw = 6;  idx = k - 10; f0 = 30 + idx * 6;   gamma = g1_gamma; beta = g1_beta; W = g1_W; bvec = g1_b; }
    else if (k < 30) { bw = 16; idx = k - 22; f0 = 102 + idx * 16; gamma = g2_gamma; beta = g2_beta; W = g2_W; bvec = g2_b; }
    else             { bw = 27; idx = 0;      f0 = 230;            gamma = g3_gamma; beta = g3_beta; W = g3_W; bvec = g3_b; }
    const int d    = bw * 2;
    const int dpad = (d + 3) & ~3;

    gamma += idx * d;
    beta  += idx * d;
    W     += (size_t)idx * d * C_OUT;
    bvec  += idx * C_OUT;

    // ---- stage weights: element (kk, n) -> W_lds[(kk/2)*288 + 2n + (kk&1)]
    //      so a B fragment {(kk,n),(kk+1,n)} is one aligned ds_load_b64 ----
    for (int i = tid; i < dpad * C_OUT; i += 256) {
        const int kk = i >> 7;
        const int n  = i & (C_OUT - 1);
        const float v = (kk < d) ? W[kk * C_OUT + n] : 0.0f;
        W_lds[(kk >> 1) * WPAIR_STRIDE + n * 2 + (kk & 1)] = v;
    }
    if (tid < DPAD_MAX) {
        gamma_lds[tid] = (tid < d) ? gamma[tid] : 1.0f;
        beta_lds[tid]  = (tid < d) ? beta[tid]  : 0.0f;
    }
    if (tid < C_OUT) bias_lds[tid] = bvec[tid];

    // ---- stage activations: per column (bin j, re/im c) the 128 time samples
    //      are contiguous in x (B,2,F,T) -> coalesced global loads ----
    for (int i = tid; i < TBLK * dpad; i += 256) {
        const int col = i >> 7;          // 0..dpad-1
        const int row = i & (TBLK - 1);  // time within tile
        float v = 0.0f;
        if (col < d) {
            const int j = col >> 1, c = col & 1;
            const int t = t_base + row;
            if (t < T_LEN)
                v = x[(((size_t)bb * 2 + c) * F_BINS + (f0 + j)) * T_LEN + t];
        }
        A_lds[row * A_STRIDE + col] = v;
    }
    __syncthreads();

    // ---- LayerNorm stats: one thread per row, two-pass (matches reference) ----
    if (tid < TBLK) {
        const float* rowp = &A_lds[tid * A_STRIDE];
        float s = 0.0f;
        for (int col = 0; col < d; ++col) s += rowp[col];
        const float mu = s / (float)d;
        float vs = 0.0f;
        for (int col = 0; col < d; ++col) {
            const float df = rowp[col] - mu;
            vs += df * df;
        }
        mu_lds[tid]   = mu;
        rstd_lds[tid] = __frsqrt_rn(vs / (float)d + LN_EPS);
    }
    __syncthreads();

    // ---- normalize in place (padded cols stay 0) ----
    for (int i = tid; i < TBLK * dpad; i += 256) {
        const int col = i >> 7;
        const int row = i & (TBLK - 1);
        if (col < d) {
            const int a = row * A_STRIDE + col;
            A_lds[a] = (A_lds[a] - mu_lds[row]) * rstd_lds[row] * gamma_lds[col] + beta_lds[col];
        }
    }
    __syncthreads();

    // ---- WMMA: each wave32 computes a 16(M) x 128(N) tile; K in steps of 4.
    //      EXEC is all-ones here (no divergence before stores). ----
    const int wave    = tid >> 5;
    const int lane    = tid & 31;
    const int lane_lo = lane & 15;
    const int lane_hi = lane >> 4;      // 0: K pair {0,1}; 1: K pair {2,3}
    const int m0      = wave * 16;      // 8 waves x 16 rows = 128 rows

    v8f acc[8] = {};

    for (int kk = 0; kk < dpad; kk += 4) {
        // A fragment (16x4 f32, ISA 7.12.2): lane_lo = M, lane_hi selects K pair
        const v2f a = *(const v2f*)&A_lds[(m0 + lane_lo) * A_STRIDE + kk + lane_hi * 2];
        const int pair_base = ((kk >> 1) + lane_hi) * WPAIR_STRIDE + lane_lo * 2;
#pragma unroll
        for (int nt = 0; nt < 8; ++nt) {
            // B fragment (4x16 f32): one aligned ds_load_b64 per lane
            const v2f bf = *(const v2f*)&W_lds[pair_base + nt * 32];
            acc[nt] = __builtin_amdgcn_wmma_f32_16x16x4_f32(
                false, a, false, bf, (short)0, acc[nt], false, false);
        }
    }

    // ---- epilogue: bias add + coalesced stores (16 lanes contiguous in N) ----
    float* outp = out + ((size_t)bb * NBANDS + k) * (size_t)T_LEN * C_OUT;
#pragma unroll
    for (int nt = 0; nt < 8; ++nt) {
        const int n = nt * 16 + lane_lo;
        const float bias = bias_lds[n];
#pragma unroll
        for (int r = 0; r < 8; ++r) {
            const int m = m0 + r + lane_hi * 8;   // C/D layout: VGPR r -> M = r (+8 for hi lanes)
            const int t = t_base + m;
            if (t < T_LEN)
                outp[(size_t)t * C_OUT + n] = acc[nt][r] + bias;
        }
    }
}

extern "C" void kernel_launch(void* const* d_in, const int* in_sizes, int n_in,
                              void* d_out, int out_size, void* d_ws, size_t ws_size,
                              hipStream_t stream) {
    (void)in_sizes; (void)n_in; (void)out_size; (void)d_ws; (void)ws_size;
    const float* x = (const float*)d_in[0];
    dim3 grid(NBANDS, (T_LEN + TBLK - 1) / TBLK, 8);
    bandsplit_wmma_kernel<<<grid, 256, 0, stream>>>(
        x,
        (const float*)d_in[1],  (const float*)d_in[2],  (const float*)d_in[3],  (const float*)d_in[4],
        (const float*)d_in[5],  (const float*)d_in[6],  (const float*)d_in[7],  (const float*)d_in[8],
        (const float*)d_in[9],  (const float*)d_in[10], (const float*)d_in[11], (const float*)d_in[12],
        (const float*)d_in[13], (const float*)d_in[14], (const float*)d_in[15], (const float*)d_in[16],
        (float*)d_out);
}